// ForceAggregation_49761491091502
// MI455X (gfx1250) — compile-verified
//
#include <hip/hip_runtime.h>

// Batched fp32 matvec f_m = H_m @ x_m via V_WMMA_F32_16X16X4_F32 (CDNA5 / gfx1250).
// Bandwidth-bound: 737 MB hess stream, ~32 us floor at 23.3 TB/s.

typedef __attribute__((ext_vector_type(2))) float v2f;
typedef __attribute__((ext_vector_type(8))) float v8f;

#define DDIM   300              // 3 * N_AT degrees of freedom
#define TILES  19               // ceil(300 / 16) row tiles per molecule
#define BLOCKT 640              // 20 wave32; waves 0..18 compute, wave 19 loads nothing (only tids < 300 stage x)

__global__ __launch_bounds__(BLOCKT)
void force_agg_wmma_kernel(const float* __restrict__ ns,    // [M, 300] newton steps (flattened)
                           const float* __restrict__ hess,  // [M*300, 300] row-stacked inverse hessians
                           float* __restrict__ out)         // [M, 300] forces (flattened)
{
    __shared__ float xs[DDIM + 4];

    const int m    = blockIdx.x;
    const int tid  = threadIdx.x;
    const int wave = tid >> 5;
    const int lane = tid & 31;

    // Stage x_m into LDS (coalesced once per block).
    const float* __restrict__ x = ns + (size_t)m * DDIM;
    for (int i = tid; i < DDIM; i += BLOCKT) xs[i] = x[i];
    __syncthreads();

    if (wave >= TILES) return;   // wave 19 idles after staging

    const int row0  = wave * 16;
    const int myrow = row0 + (lane & 15);
    const int rowc  = (myrow < DDIM) ? myrow : (DDIM - 1);   // clamp last partial tile (no OOB, no divergence)
    const int khalf = (lane < 16) ? 0 : 2;                   // A layout: V0/V1 hold K={0,1} lanes 0-15, K={2,3} lanes 16-31

    const float* __restrict__ hrow = hess + ((size_t)m * DDIM + rowc) * DDIM + khalf;

    v8f acc = {};
    #pragma unroll 5
    for (int k0 = 0; k0 < DDIM; k0 += 4) {
        // A: 16x4 fp32 hess tile, one b64 load per lane (256B of hess per wmma).
        v2f a = *(const v2f*)(hrow + k0);
        // B: x segment broadcast across all 16 output columns (half-wave K split mirrors A).
        v2f b;
        b.x = xs[k0 + khalf];
        b.y = xs[k0 + khalf + 1];
        // D = A*B + C : (neg_a, A, neg_b, B, c_mod, C, reuse_a, reuse_b)
        acc = __builtin_amdgcn_wmma_f32_16x16x4_f32(false, a, false, b,
                                                    (short)0, acc, false, false);
    }

    // C/D layout: lane L, VGPR v -> row = v + (L<16 ? 0 : 8), col = L&15.
    // Every column holds the same matvec result; column 0 lives in lanes 0 and 16.
    if ((lane & 15) == 0) {
        const int rbase = (lane < 16) ? 0 : 8;
        float* __restrict__ o = out + (size_t)m * DDIM;
        #pragma unroll
        for (int v = 0; v < 8; ++v) {
            const int r = row0 + rbase + v;
            if (r < DDIM) o[r] = acc[v];
        }
    }
}

extern "C" void kernel_launch(void* const* d_in, const int* in_sizes, int n_in,
                              void* d_out, int out_size, void* d_ws, size_t ws_size,
                              hipStream_t stream) {
    const float* ns   = (const float*)d_in[0];   // [M*N_AT, 3] == [M, 300]
    const float* hess = (const float*)d_in[1];   // [M*300, 300]
    // d_in[2] (idx_m) and d_in[3] (n_atoms) are implied by the uniform layout.
    float* out = (float*)d_out;                  // [M*N_AT, 3] == [M, 300]

    const int M = in_sizes[3];                   // n_atoms has one entry per molecule
    dim3 grid(M), block(BLOCKT);
    hipLaunchKernelGGL(force_agg_wmma_kernel, grid, block, 0, stream, ns, hess, out);
}